// GaussianSplatting_76295799046180
// MI455X (gfx1250) — compile-verified
//
#include <hip/hip_runtime.h>
#include <stdint.h>

// Gaussian-splatting preprocess for MI455X (gfx1250).
// Roofline: ~288 B/point * 2M points = 576 MB @ 23.3 TB/s => ~25 us floor; ~0.6 GFLOP
// total compute => purely HBM-bound. All heavy traffic goes through the CDNA5 async
// global<->LDS path (ASYNCcnt): features_rest staged in coalesced 16B chunks on the
// way in, all four output sections drained with coalesced 16B chunks on the way out.

#define TPB   256
#define RESTF 45                         // 15 SH coeffs * 3 channels
#define OUTF  13                         // 3 pos + 1 opac + 3 color + 6 cov

// LDS float offsets of the output staging sections
#define SO_PTS  0                        // 3*TPB floats
#define SO_OPA  (3 * TPB)                // TPB floats
#define SO_COL  (4 * TPB)                // 3*TPB floats
#define SO_COV  (7 * TPB)                // 6*TPB floats

// Cooperative async LDS->global copy of cnt_f floats. lds_byte is 16B aligned by
// construction; g_byte is 16B aligned for the actual problem shape (N % 4 == 0),
// with a b32 fallback otherwise. All threads of the block participate.
__device__ __forceinline__ void async_store_section(
    unsigned lds_byte, unsigned g_byte, unsigned cnt_f, uint64_t gbase, int tid)
{
    if ((g_byte & 15u) == 0u) {
        const unsigned chunks = cnt_f >> 2;
        const unsigned rem    = cnt_f & 3u;
        for (unsigned i = (unsigned)tid; i < chunks; i += TPB) {
            const unsigned b = i * 16u;
            asm volatile("global_store_async_from_lds_b128 %0, %1, %2"
                         :: "v"(g_byte + b), "v"(lds_byte + b), "s"(gbase)
                         : "memory");
        }
        if ((unsigned)tid < rem) {
            const unsigned b = chunks * 16u + (unsigned)tid * 4u;
            asm volatile("global_store_async_from_lds_b32 %0, %1, %2"
                         :: "v"(g_byte + b), "v"(lds_byte + b), "s"(gbase)
                         : "memory");
        }
    } else {
        for (unsigned i = (unsigned)tid; i < cnt_f; i += TPB) {
            const unsigned b = i * 4u;
            asm volatile("global_store_async_from_lds_b32 %0, %1, %2"
                         :: "v"(g_byte + b), "v"(lds_byte + b), "s"(gbase)
                         : "memory");
        }
    }
}

__global__ __launch_bounds__(TPB) void gs_preprocess(
    const float* __restrict__ xyz,
    const float* __restrict__ f_dc,
    const float* __restrict__ f_rest,
    const float* __restrict__ scaling,
    const float* __restrict__ rotation,
    const float* __restrict__ opacity,
    const float* __restrict__ campos,
    const float* __restrict__ smod,
    float* __restrict__ out,
    int N)
{
    __shared__ float s_rest[TPB * RESTF];   // 46080 B : staged features_rest
    __shared__ float s_out[TPB * OUTF];     // 13312 B : staged outputs (59392 B total)

    const int tid    = (int)threadIdx.x;
    const int block0 = (int)blockIdx.x * TPB;
    int valid = N - block0; if (valid > TPB) valid = TPB;
    if (valid <= 0) return;                 // uniform per block

    // ---- Stage features_rest via CDNA5 async global->LDS (coalesced 16B chunks) ----
    {
        const uint64_t gb    = (uint64_t)(uintptr_t)f_rest;
        const unsigned gbase = (unsigned)block0 * (RESTF * 4u);      // byte offset (<4GB)
        const unsigned lds0  = (unsigned)(uintptr_t)&s_rest[0];
        const unsigned totalFloats = (unsigned)valid * RESTF;
        const unsigned nChunks     = totalFloats >> 2;
        const unsigned rem         = totalFloats & 3u;

        for (unsigned i = (unsigned)tid; i < nChunks; i += TPB) {
            const unsigned b = i * 16u;
            asm volatile("global_load_async_to_lds_b128 %0, %1, %2"
                         :: "v"(lds0 + b), "v"(gbase + b), "s"(gb)
                         : "memory");
        }
        if ((unsigned)tid < rem) {
            const unsigned b = nChunks * 16u + (unsigned)tid * 4u;
            asm volatile("global_load_async_to_lds_b32 %0, %1, %2"
                         :: "v"(lds0 + b), "v"(gbase + b), "s"(gb)
                         : "memory");
        }
        asm volatile("s_wait_asynccnt 0x0" ::: "memory");
    }
    __syncthreads();

    // ---- Per-point compute (tail threads idle but stay for the barrier) ----
    if (tid < valid) {
        const int p = block0 + tid;

        const float px = xyz[3 * p + 0];
        const float py = xyz[3 * p + 1];
        const float pz = xyz[3 * p + 2];
        const float mod = smod[0];

        // view direction
        float dx = px - campos[0];
        float dy = py - campos[1];
        float dz = pz - campos[2];
        const float dinv = rsqrtf(dx * dx + dy * dy + dz * dz);
        const float x = dx * dinv, y = dy * dinv, z = dz * dinv;
        const float xx = x * x, yy = y * y, zz = z * z;
        const float xy = x * y, yz = y * z, xz = x * z;

        // SH degree-3 evaluation; coefficients from LDS (stride 45 => conflict-free)
        const float* rest = &s_rest[tid * RESTF];
        float cr, cg, cb;
        {
            const float w0 = 0.28209479177387814f;
            cr = fmaf(w0, f_dc[3 * p + 0], 0.5f);
            cg = fmaf(w0, f_dc[3 * p + 1], 0.5f);
            cb = fmaf(w0, f_dc[3 * p + 2], 0.5f);
        }
#define ACC(K, W) { const float w_ = (W);                      \
        cr = fmaf(w_, rest[(K) * 3 + 0], cr);                  \
        cg = fmaf(w_, rest[(K) * 3 + 1], cg);                  \
        cb = fmaf(w_, rest[(K) * 3 + 2], cb); }
        ACC(0,  -0.4886025119029199f * y)
        ACC(1,   0.4886025119029199f * z)
        ACC(2,  -0.4886025119029199f * x)
        ACC(3,   1.0925484305920792f * xy)
        ACC(4,  -1.0925484305920792f * yz)
        ACC(5,   0.31539156525252005f * (2.0f * zz - xx - yy))
        ACC(6,  -1.0925484305920792f * xz)
        ACC(7,   0.5462742152960396f * (xx - yy))
        ACC(8,  -0.5900435899266435f * y * (3.0f * xx - yy))
        ACC(9,   2.890611442640554f  * xy * z)
        ACC(10, -0.4570457994644658f * y * (4.0f * zz - xx - yy))
        ACC(11,  0.3731763325901154f * z * (2.0f * zz - 3.0f * xx - 3.0f * yy))
        ACC(12, -0.4570457994644658f * x * (4.0f * zz - xx - yy))
        ACC(13,  1.445305721320277f  * z * (xx - yy))
        ACC(14, -0.5900435899266435f * x * (xx - 3.0f * yy))
#undef ACC
        cr = fmaxf(cr, 0.0f);
        cg = fmaxf(cg, 0.0f);
        cb = fmaxf(cb, 0.0f);

        // opacity sigmoid
        const float opac = 1.0f / (1.0f + expf(-opacity[p]));

        // covariance from scaling + rotation (rotation is 16B-aligned: one b128 load)
        const float sx = expf(scaling[3 * p + 0]) * mod;
        const float sy = expf(scaling[3 * p + 1]) * mod;
        const float sz = expf(scaling[3 * p + 2]) * mod;

        const float4 q4 = ((const float4*)rotation)[p];
        float qw = q4.x, qx = q4.y, qy = q4.z, qz = q4.w;
        const float qinv = rsqrtf(qw * qw + qx * qx + qy * qy + qz * qz);
        qw *= qinv; qx *= qinv; qy *= qinv; qz *= qinv;

        const float r00 = 1.0f - 2.0f * (qy * qy + qz * qz);
        const float r01 = 2.0f * (qx * qy - qw * qz);
        const float r02 = 2.0f * (qx * qz + qw * qy);
        const float r10 = 2.0f * (qx * qy + qw * qz);
        const float r11 = 1.0f - 2.0f * (qx * qx + qz * qz);
        const float r12 = 2.0f * (qy * qz - qw * qx);
        const float r20 = 2.0f * (qx * qz - qw * qy);
        const float r21 = 2.0f * (qy * qz + qw * qx);
        const float r22 = 1.0f - 2.0f * (qx * qx + qy * qy);

        const float l00 = r00 * sx, l01 = r01 * sy, l02 = r02 * sz;
        const float l10 = r10 * sx, l11 = r11 * sy, l12 = r12 * sz;
        const float l20 = r20 * sx, l21 = r21 * sy, l22 = r22 * sz;

        // ---- stage all 13 outputs in LDS (strides 3/1/3/6 => conflict-free) ----
        s_out[SO_PTS + 3 * tid + 0] = px;
        s_out[SO_PTS + 3 * tid + 1] = py;
        s_out[SO_PTS + 3 * tid + 2] = pz;

        s_out[SO_OPA + tid] = opac;

        s_out[SO_COL + 3 * tid + 0] = cr;
        s_out[SO_COL + 3 * tid + 1] = cg;
        s_out[SO_COL + 3 * tid + 2] = cb;

        s_out[SO_COV + 6 * tid + 0] = l00 * l00 + l01 * l01 + l02 * l02;
        s_out[SO_COV + 6 * tid + 1] = l00 * l10 + l01 * l11 + l02 * l12;
        s_out[SO_COV + 6 * tid + 2] = l00 * l20 + l01 * l21 + l02 * l22;
        s_out[SO_COV + 6 * tid + 3] = l10 * l10 + l11 * l11 + l12 * l12;
        s_out[SO_COV + 6 * tid + 4] = l10 * l20 + l11 * l21 + l12 * l22;
        s_out[SO_COV + 6 * tid + 5] = l20 * l20 + l21 * l21 + l22 * l22;
    }
    __syncthreads();

    // ---- Drain outputs: cooperative async LDS->global, per contiguous section ----
    {
        const uint64_t gb   = (uint64_t)(uintptr_t)out;
        const unsigned so   = (unsigned)(uintptr_t)&s_out[0];   // LDS byte base
        const unsigned uN   = (unsigned)N;
        const unsigned ub0  = (unsigned)block0;
        const unsigned v    = (unsigned)valid;

        // points | opac | colors | covariance (each contiguous per block in global)
        async_store_section(so + SO_PTS * 4u, (3u * ub0)            * 4u, 3u * v, gb, tid);
        async_store_section(so + SO_OPA * 4u, (3u * uN + ub0)       * 4u,      v, gb, tid);
        async_store_section(so + SO_COL * 4u, (4u * uN + 3u * ub0)  * 4u, 3u * v, gb, tid);
        async_store_section(so + SO_COV * 4u, (7u * uN + 6u * ub0)  * 4u, 6u * v, gb, tid);

        asm volatile("s_wait_asynccnt 0x0" ::: "memory");
    }
}

extern "C" void kernel_launch(void* const* d_in, const int* in_sizes, int n_in,
                              void* d_out, int out_size, void* d_ws, size_t ws_size,
                              hipStream_t stream) {
    const float* xyz      = (const float*)d_in[0];
    const float* f_dc     = (const float*)d_in[1];
    const float* f_rest   = (const float*)d_in[2];
    const float* scaling  = (const float*)d_in[3];
    const float* rotation = (const float*)d_in[4];
    const float* opacity  = (const float*)d_in[5];
    const float* campos   = (const float*)d_in[6];
    const float* smod     = (const float*)d_in[7];
    float* out = (float*)d_out;

    const int N = in_sizes[0] / 3;
    const int blocks = (N + TPB - 1) / TPB;

    hipLaunchKernelGGL(gs_preprocess, dim3(blocks), dim3(TPB), 0, stream,
                       xyz, f_dc, f_rest, scaling, rotation, opacity,
                       campos, smod, out, N);
}